// IntuitionFields_764504179011
// MI455X (gfx1250) — compile-verified
//
#include <hip/hip_runtime.h>
#include <stdint.h>

#define N_BOX 128
#define HW (512*512)
#define MTILES 8
#define KSLICES 128
#define SLICE_PIX 2048
#define CHUNK 512
#define NCHUNK (SLICE_PIX / CHUNK)
#define PITCH 516           // CHUNK + 4 dwords: bank-conflict-free b64 A reads

typedef float v2f __attribute__((ext_vector_type(2)));
typedef float v8f __attribute__((ext_vector_type(8)));
typedef int   v4i __attribute__((ext_vector_type(4)));

typedef __attribute__((address_space(1))) v4i* g_v4i_p;
typedef __attribute__((address_space(3))) v4i* l_v4i_p;

// ---------------- CDNA5 async global->LDS copy (ASYNCcnt path) ----------------
__device__ __forceinline__ void async_ld_b128(const float* g, float* l) {
#if __has_builtin(__builtin_amdgcn_global_load_async_to_lds_b128)
  __builtin_amdgcn_global_load_async_to_lds_b128(
      (g_v4i_p)(void*)g, (l_v4i_p)(void*)l, 0, 0);
#else
  // generic LDS pointer: low 32 bits are the LDS byte offset (ISA 10.2)
  asm volatile("global_load_async_to_lds_b128 %0, %1, off"
               :: "v"((unsigned)(uintptr_t)l),
                  "v"((unsigned long long)(uintptr_t)g)
               : "memory");
#endif
}

__device__ __forceinline__ void wait_async0() {
#if __has_builtin(__builtin_amdgcn_s_wait_asynccnt)
  __builtin_amdgcn_s_wait_asynccnt(0);
#else
  asm volatile("s_wait_asynccnt 0" ::: "memory");
#endif
}

__device__ __forceinline__ float sigm(float x) { return 1.0f / (1.0f + __expf(-x)); }
__device__ __forceinline__ float clamp01(float x) { return fminf(fmaxf(x, 0.0f), 1.0f); }

// ======================= Kernel 1: masked moments as WMMA GEMM =======================
// D(16 masks x 3 stats) += bin(mask)(16xK) x stats(Kx3), K accumulated 4 at a time.
__global__ __launch_bounds__(256) void k_moments(const float* __restrict__ masks,
                                                 const float* __restrict__ depth,
                                                 float* __restrict__ partials) {
  __shared__ float mbuf[2][16 * PITCH];
  __shared__ float dbuf[2][CHUNK];
  __shared__ float plds[8 * 64];

  const int mtile = blockIdx.x;
  const int slice = blockIdx.y;
  const int t    = threadIdx.x;       // 256 threads = 8 waves
  const int wave = t >> 5;
  const int lane = t & 31;
  const int nn   = lane & 15;   // N (stat column) for B/C/D; M row for A
  const int hi   = lane >> 4;
  const int pix_base = slice * SLICE_PIX;
  const float* mbase = masks + (size_t)mtile * 16 * HW + pix_base;

  // one-hot per-lane column weights: stat(n,d) = dv * (w0 + d*(w1 + d*w2))
  const float w0 = (nn == 0) ? 1.0f : 0.0f;
  const float w1 = (nn == 1) ? 1.0f : 0.0f;
  const float w2 = (nn == 2) ? 1.0f : 0.0f;

  auto issue = [&](int c) {
    const int b = c & 1;
    const int pp = c * CHUNK;
    const int c4 = (t & 127) * 4;
    const int rh = (t >> 7);          // 0 or 1
#pragma unroll
    for (int i = 0; i < 8; ++i) {     // 16 masks x 512 px, coalesced 16B/lane
      const int r = i * 2 + rh;
      async_ld_b128(mbase + (size_t)r * HW + pp + c4, &mbuf[b][r * PITCH + c4]);
    }
    if (t < 128) async_ld_b128(depth + pix_base + pp + t * 4, &dbuf[b][t * 4]);
  };

  v8f acc = {};
  issue(0);
  for (int c = 0; c < NCHUNK; ++c) {
    wait_async0();          // own chunk-c loads landed in LDS
    __syncthreads();        // everyone's landed; prior readers of other buffer done
    if (c + 1 < NCHUNK) issue(c + 1);   // overlap next loads with this compute
    const int b = c & 1;
    const float* mrow = &mbuf[b][nn * PITCH];
    const int kw = wave * 64;           // this wave's 64-pixel window of the chunk
#pragma unroll 4
    for (int kk = 0; kk < 64; kk += 4) {
      const int k0 = kw + kk + 2 * hi;  // A/B: vgpr j, lane-half h -> K = 2h + j
      const float2 mv = *(const float2*)&mrow[k0];
      v2f a;
      a.x = mv.x > 0.5f ? 1.0f : 0.0f;
      a.y = mv.y > 0.5f ? 1.0f : 0.0f;
      const float2 dd = *(const float2*)&dbuf[b][k0];
      const float dv0 = dd.x > 1e-4f ? 1.0f : 0.0f;
      const float dv1 = dd.y > 1e-4f ? 1.0f : 0.0f;
      v2f bb;                            // branchless, EXEC untouched
      bb.x = dv0 * fmaf(dd.x, fmaf(dd.x, w2, w1), w0);
      bb.y = dv1 * fmaf(dd.y, fmaf(dd.y, w2, w1), w0);
      acc = __builtin_amdgcn_wmma_f32_16x16x4_f32(false, a, false, bb, (short)0, acc,
                                                  false, false);
    }
    __syncthreads();        // buffer b free for reuse
  }

  // Deterministic per-block combine of the 8 waves' partials, no atomics.
  plds[t] = 0.0f; plds[t + 256] = 0.0f;
  __syncthreads();
  if (nn < 3) {
#pragma unroll
    for (int r = 0; r < 8; ++r)       // D: vgpr r, lane-half hi -> M = r + 8*hi
      plds[wave * 64 + (r + 8 * hi) * 4 + nn] = acc[r];
  }
  __syncthreads();
  if (t < 64) {
    float s = 0.0f;
#pragma unroll
    for (int w = 0; w < 8; ++w) s += plds[w * 64 + t];
    partials[((size_t)mtile * KSLICES + slice) * 64 + t] = s;
  }
}

// ======================= Kernel 2: finalize per-mask fields =======================
__global__ __launch_bounds__(128) void k_finalize(const float* __restrict__ boxes,
                                                  const float* __restrict__ conf,
                                                  const float* __restrict__ partials,
                                                  float* __restrict__ out) {
  const int i = threadIdx.x;          // one block of 128
  const int mtile = i >> 4, mi = i & 15;
  float cnt = 0.0f, sum = 0.0f, sq = 0.0f;
  for (int s = 0; s < KSLICES; ++s) { // fixed-order deterministic reduction
    const float* p = &partials[((size_t)mtile * KSLICES + s) * 64 + mi * 4];
    cnt += p[0]; sum += p[1]; sq += p[2];
  }
  const float cntf = fmaxf(cnt, 1.0f);
  const float mean = sum / cntf;
  float var = (sq - 2.0f * mean * sum + mean * mean * cnt) / cntf;
  var = (cnt > 0.0f) ? fmaxf(var, 0.0f) : 0.0f;

  const float x1 = boxes[i * 4 + 0], y1 = boxes[i * 4 + 1];
  const float x2 = boxes[i * 4 + 2], y2 = boxes[i * 4 + 3];
  const float bw = fmaxf(x2 - x1, 0.0f), bh = fmaxf(y2 - y1, 0.0f);
  const float area = clamp01(bw * bh);
  const float tall = fminf(fmaxf(bh / (bw + 1e-6f), 0.0f), 10.0f);
  const float thin = sigm((tall - 1.2f) * 2.0f);
  const float cf = conf[i];
  const float unstable = clamp01(0.35f * thin + 0.35f * sigm(var * 6.0f) + 0.3f * (1.0f - cf));
  const float slip = clamp01(0.45f * sigm(var * 8.0f) + 0.25f * (1.0f - cf) +
                             0.3f * sigm((area - 0.05f) * 3.0f));
  const float support = clamp01(1.0f - unstable);
  out[i] = unstable;
  out[128 + i] = slip;
  out[256 + i] = support;
}

// ======================= Kernel 3: IoU + exact top-64 =======================
__global__ __launch_bounds__(256) void k_iou_topk(const float* __restrict__ boxes,
                                                  float* __restrict__ out) {
  __shared__ float iou_s[N_BOX * N_BOX];   // 64 KB (CDNA5 WGP has 320 KB LDS)
  __shared__ float bx[N_BOX * 4];
  __shared__ float ar[N_BOX];
  __shared__ unsigned long long red[256];
  const int t = threadIdx.x;

  for (int i = t; i < N_BOX * 4; i += 256) bx[i] = boxes[i];
  __syncthreads();
  for (int i = t; i < N_BOX; i += 256) {
    const float bw = fmaxf(bx[i * 4 + 2] - bx[i * 4 + 0], 0.0f);
    const float bh = fmaxf(bx[i * 4 + 3] - bx[i * 4 + 1], 0.0f);
    ar[i] = bw * bh;
  }
  __syncthreads();
  for (int p = t; p < N_BOX * N_BOX; p += 256) {
    const int i = p >> 7, j = p & 127;
    float v = 0.0f;
    if (i != j) {
      const float xx1 = fmaxf(bx[i * 4 + 0], bx[j * 4 + 0]);
      const float yy1 = fmaxf(bx[i * 4 + 1], bx[j * 4 + 1]);
      const float xx2 = fminf(bx[i * 4 + 2], bx[j * 4 + 2]);
      const float yy2 = fminf(bx[i * 4 + 3], bx[j * 4 + 3]);
      const float inter = fmaxf(xx2 - xx1, 0.0f) * fmaxf(yy2 - yy1, 0.0f);
      v = inter / (ar[i] + ar[j] - inter + 1e-6f);
    }
    iou_s[p] = v;
  }
  __syncthreads();

  int* pout = (int*)out + 384;   // pairs: int32 bits at float-offset 384..511
  float* cout = out + 512;       // collision: 512..575
  for (int r = 0; r < 64; ++r) {
    unsigned long long best = 0ull;
    for (int p = t; p < N_BOX * N_BOX; p += 256) {
      unsigned ub = __float_as_uint(iou_s[p]);
      ub ^= (unsigned)(((int)ub >> 31)) | 0x80000000u;  // order-preserving encode
      const unsigned long long key =
          ((unsigned long long)ub << 32) | (unsigned)(0xFFFFFFFFu - p);
      best = key > best ? key : best;                   // max val, tie -> min idx
    }
    red[t] = best;
    __syncthreads();
    for (int s2 = 128; s2 > 0; s2 >>= 1) {
      if (t < s2) red[t] = red[t] > red[t + s2] ? red[t] : red[t + s2];
      __syncthreads();
    }
    if (t == 0) {
      const unsigned long long k = red[0];
      const int idx = (int)(0xFFFFFFFFu - (unsigned)(k & 0xFFFFFFFFu));
      const unsigned ub = (unsigned)(k >> 32);
      const unsigned fb = (ub & 0x80000000u) ? (ub ^ 0x80000000u) : ~ub;
      const float v = __uint_as_float(fb);
      const bool valid = v > 0.02f;
      pout[r * 2 + 0] = valid ? (idx >> 7) : 0;
      pout[r * 2 + 1] = valid ? (idx & 127) : 0;
      cout[r] = valid ? clamp01(v * 5.0f) : 0.0f;
      iou_s[idx] = -2.0f;   // exclude from later rounds
    }
    __syncthreads();
  }
}

extern "C" void kernel_launch(void* const* d_in, const int* in_sizes, int n_in,
                              void* d_out, int out_size, void* d_ws, size_t ws_size,
                              hipStream_t stream) {
  const float* boxes = (const float*)d_in[0];
  const float* masks = (const float*)d_in[1];
  const float* conf  = (const float*)d_in[2];
  const float* depth = (const float*)d_in[3];
  float* out = (float*)d_out;
  float* partials = (float*)d_ws;   // 8*128*64 floats = 256 KB scratch

  k_moments<<<dim3(MTILES, KSLICES), 256, 0, stream>>>(masks, depth, partials);
  k_finalize<<<1, 128, 0, stream>>>(boxes, conf, partials, out);
  k_iou_topk<<<1, 256, 0, stream>>>(boxes, out);
}